// WSGATWrapper_83330955477485
// MI455X (gfx1250) — compile-verified
//
#include <hip/hip_runtime.h>
#include <hip/hip_bf16.h>
#include <math.h>

typedef __attribute__((ext_vector_type(2))) float v2f;
typedef __attribute__((ext_vector_type(8))) float v8f;

#define HEADS 4
#define CDIM  32
#define DDIM  128   // H*C
#define KC    64    // K chunk staged in LDS
#define KPAD  (KC + 2)   // LDS row stride (floats): keeps 8B alignment, bank step 2

// ---------------------------------------------------------------------------
// WMMA f32 GEMM:  C[M x 128] = gather(A)[M x K] @ B[K x 128] (+ bias[128])
// M%16==0, K%64==0, Nc==128. Block = 256 thr = 8 waves; block b owns M-tile b,
// wave w owns 16-col N-tile w. A chunk (16xKC) and B chunk (KCx128, stored
// transposed) are staged cooperatively in LDS, then each wave runs
// V_WMMA_F32_16X16X4_F32 fed by two ds_load_b64 per k-step.
// Frag layouts (ISA 7.12.2): A 16x4: lane = m + 16*(k>=2), vgpr = k&1 (B mirrored);
// C/D: vgpr v -> m = v + 8*(lane>=16), n = lane&15.
// ---------------------------------------------------------------------------
__global__ void gemm_wmma_f32(const float* __restrict__ A, const float* __restrict__ B,
                              const float* __restrict__ bias, const int* __restrict__ rowIdx,
                              float* __restrict__ C, int M, int K, int Nc) {
  __shared__ float As[16 * KPAD];
  __shared__ float Bs[128 * KPAD];

  const int wave   = threadIdx.x >> 5;
  const int lane   = threadIdx.x & 31;
  const int laneLo = lane & 15;
  const int laneHi = lane >> 4;          // 0 or 1
  const int mTile  = blockIdx.x;
  const int col0   = wave * 16;

  v8f acc = {0.f, 0.f, 0.f, 0.f, 0.f, 0.f, 0.f, 0.f};

  for (int k0 = 0; k0 < K; k0 += KC) {
    // stage A: rows mTile*16 .. +15, cols k0 .. k0+KC-1 (gathered via rowIdx)
    for (int idx = threadIdx.x; idx < 16 * KC; idx += 256) {
      int r = idx / KC, c = idx - r * KC;
      int row  = mTile * 16 + r;
      int arow = rowIdx ? rowIdx[row] : row;
      As[r * KPAD + c] = A[(size_t)arow * K + k0 + c];
    }
    // stage B transposed: Bs[c][r] = B[k0+r][c]; coalesced global reads,
    // LDS writes stride KPAD floats -> bank step 2 (conflict-free)
    for (int idx = threadIdx.x; idx < KC * 128; idx += 256) {
      int r = idx >> 7;        // 0..KC-1
      int c = idx & 127;       // 0..127
      Bs[c * KPAD + r] = B[(size_t)(k0 + r) * Nc + c];
    }
    __syncthreads();
#pragma unroll
    for (int k = 0; k < KC; k += 4) {
      v2f af = *(const v2f*)(As + laneLo * KPAD + k + 2 * laneHi);
      v2f bf = *(const v2f*)(Bs + (col0 + laneLo) * KPAD + k + 2 * laneHi);
      acc = __builtin_amdgcn_wmma_f32_16x16x4_f32(false, af, false, bf,
                                                  (short)0, acc, false, false);
    }
    __syncthreads();
  }

#pragma unroll
  for (int v = 0; v < 8; ++v) {
    const int m = v + 8 * laneHi;
    const int n = col0 + laneLo;
    float val = acc[v];
    if (bias) val += bias[n];
    C[(size_t)(mTile * 16 + m) * Nc + n] = val;
  }
}

// ---------------------------------------------------------------------------
// si[n,h] = xh[n,h,:]·wi[h,:] ; sj[n,h] = xh[n,h,:]·wj[h,:]
// att_l layout: (H, 1, 3C): wi = att[h][0:32], wj = att[h][32:64], we = att[h][64:96]
// ---------------------------------------------------------------------------
__global__ void node_scores(const float* __restrict__ xh, const float* __restrict__ att_l,
                            float* __restrict__ si, float* __restrict__ sj, int Nn) {
  int idx = blockIdx.x * blockDim.x + threadIdx.x;
  if (idx >= Nn * HEADS) return;
  int n = idx >> 2, h = idx & 3;
  const float* xp = xh + (size_t)n * DDIM + h * CDIM;
  const float* wi = att_l + h * (3 * CDIM);
  const float* wj = wi + CDIM;
  float a = 0.f, b = 0.f;
#pragma unroll
  for (int c = 0; c < CDIM; ++c) { float v = xp[c]; a += v * wi[c]; b += v * wj[c]; }
  si[idx] = a;
  sj[idx] = b;
}

// ---------------------------------------------------------------------------
// k[h]  = ((W_ew @ W_edge_l)|head h) · we_h      (rank-1 edge_attr folding)
// kb[h] = ((b_ew @ W_edge_l)|head h) · we_h
// single block of 128 threads
// ---------------------------------------------------------------------------
__global__ void edge_k(const float* __restrict__ W_ew, const float* __restrict__ b_ew,
                       const float* __restrict__ W_edge_l, const float* __restrict__ att_l,
                       float* __restrict__ kout /* [8]: k[4], kb[4] */) {
  __shared__ float v[DDIM], vb[DDIM];
  int j = threadIdx.x;
  float s = 0.f, sb = 0.f;
  for (int d = 0; d < DDIM; ++d) {
    float wcol = W_edge_l[(size_t)d * DDIM + j];
    s  += W_ew[d] * wcol;
    sb += b_ew[d] * wcol;
  }
  v[j] = s; vb[j] = sb;
  __syncthreads();
  if (j < HEADS) {
    const float* we = att_l + j * (3 * CDIM) + 2 * CDIM;
    float k = 0.f, kb = 0.f;
#pragma unroll
    for (int c = 0; c < CDIM; ++c) { k += v[j * CDIM + c] * we[c]; kb += vb[j * CDIM + c] * we[c]; }
    kout[j] = k;
    kout[HEADS + j] = kb;
  }
}

__global__ void zero_f32(float* p, long long n) {
  long long i = (long long)blockIdx.x * blockDim.x + threadIdx.x;
  if (i < n) p[i] = 0.f;
}

// ---------------------------------------------------------------------------
// Pass A: a[e,h] = tanh(si[d,h] + sj[s,h] + se); atomicMax |a| into m[d,h]
// Edges e < E are real edges; e in [E, E+N) are self loops (se = 0).
// |tanh| >= 0, so uint-bit atomicMax against 0-initialized m is exact.
// ---------------------------------------------------------------------------
__global__ void edge_score(const int* __restrict__ src, const int* __restrict__ dst,
                           const float* __restrict__ ew, const float* __restrict__ si,
                           const float* __restrict__ sj, const float* __restrict__ kh,
                           float* __restrict__ aout, unsigned int* __restrict__ mmax,
                           int E_, int Ntot) {
  int e = blockIdx.x * blockDim.x + threadIdx.x;
  if (e >= Ntot) return;
  int s, d; float w; bool real = (e < E_);
  if (real) { s = src[e]; d = dst[e]; w = ew[e]; }
  else      { s = d = e - E_; w = 0.f; }
#pragma unroll
  for (int h = 0; h < HEADS; ++h) {
    float se = real ? (kh[h] * w + kh[HEADS + h]) : 0.f;
    float a  = tanhf(si[d * HEADS + h] + sj[s * HEADS + h] + se);
    aout[(size_t)e * HEADS + h] = a;
    atomicMax(&mmax[d * HEADS + h], __float_as_uint(fabsf(a)));
  }
}

// Pass B: den[d,h] += exp(|a| - m[d,h])
__global__ void edge_den(const int* __restrict__ dst, const float* __restrict__ aout,
                         const float* __restrict__ mmax, float* __restrict__ den,
                         int E_, int Ntot) {
  int e = blockIdx.x * blockDim.x + threadIdx.x;
  if (e >= Ntot) return;
  int d = (e < E_) ? dst[e] : (e - E_);
#pragma unroll
  for (int h = 0; h < HEADS; ++h) {
    float aa = fabsf(aout[(size_t)e * HEADS + h]);
    atomicAdd(&den[d * HEADS + h], expf(aa - mmax[d * HEADS + h]));
  }
}

// Pass C: one wave per edge; lane covers 4 cols (head = lane>>3).
// out[d,:] += alpha[h] * xh[s,:]   via global_atomic_add_f32
__global__ void edge_aggregate(const int* __restrict__ src, const int* __restrict__ dst,
                               const float* __restrict__ aout, const float* __restrict__ mmax,
                               const float* __restrict__ den, const float* __restrict__ xh,
                               float* __restrict__ out, int E_, int Ntot) {
  long long gid = (long long)blockIdx.x * blockDim.x + threadIdx.x;
  int e = (int)(gid >> 5);
  int lane = (int)(gid & 31);
  if (e >= Ntot) return;
  int s, d;
  if (e < E_) { s = src[e]; d = dst[e]; }
  else        { s = d = e - E_; }
  int h = lane >> 3;
  float a   = aout[(size_t)e * HEADS + h];
  float ex  = expf(fabsf(a) - mmax[d * HEADS + h]);
  float sgn = (a > 0.f) ? 1.f : ((a < 0.f) ? -1.f : 0.f);
  float alpha = sgn * ex / (den[d * HEADS + h] + 1e-16f);
  const float4 v = ((const float4*)(xh + (size_t)s * DDIM))[lane];
  float* op = out + (size_t)d * DDIM + lane * 4;
  atomicAdd(op + 0, alpha * v.x);
  atomicAdd(op + 1, alpha * v.y);
  atomicAdd(op + 2, alpha * v.z);
  atomicAdd(op + 3, alpha * v.w);
}

// out = acc + conv_bias, optional ELU
__global__ void finalize(const float* __restrict__ acc, const float* __restrict__ bias,
                         float* __restrict__ xout, long long total, int elu) {
  long long i = (long long)blockIdx.x * blockDim.x + threadIdx.x;
  if (i >= total) return;
  float v = acc[i] + bias[i & (DDIM - 1)];
  if (elu) v = (v > 0.f) ? v : (expf(v) - 1.f);
  xout[i] = v;
}

// ---------------------------------------------------------------------------
extern "C" void kernel_launch(void* const* d_in, const int* in_sizes, int n_in,
                              void* d_out, int out_size, void* d_ws, size_t ws_size,
                              hipStream_t stream) {
  const int*   entity   = (const int*)  d_in[0];
  const int*   eidx     = (const int*)  d_in[1];
  const float* eweight  = (const float*)d_in[2];
  const float* emb      = (const float*)d_in[3];
  const float* W_proj   = (const float*)d_in[4];
  const float* b_proj   = (const float*)d_in[5];
  const float* W_ew     = (const float*)d_in[6];
  const float* b_ew     = (const float*)d_in[7];
  const float* W_src    = (const float*)d_in[8];
  const float* att      = (const float*)d_in[9];
  const float* W_edge   = (const float*)d_in[10];
  const float* cbias    = (const float*)d_in[11];

  const int Nn   = in_sizes[0];                 // 50000
  const int E_   = in_sizes[2];                 // 800000
  const int F_   = in_sizes[4] / DDIM;          // 256
  const int L_   = in_sizes[8] / (DDIM * DDIM); // 2
  const int Ntot = E_ + Nn;                     // edges + self loops

  // workspace carve (all f32)
  char* base = (char*)d_ws;
  size_t off = 0;
  auto carve = [&](size_t nfloats) { float* p = (float*)(base + off);
                                     off += ((nfloats * 4 + 255) & ~(size_t)255); return p; };
  float* x   = carve((size_t)Nn * DDIM);
  float* xh  = carve((size_t)Nn * DDIM);
  float* acc = carve((size_t)Nn * DDIM);
  float* aE  = carve((size_t)Ntot * HEADS);
  float* si  = carve((size_t)Nn * HEADS);
  float* sj  = carve((size_t)Nn * HEADS);
  float* mM  = carve((size_t)Nn * HEADS);
  float* den = carve((size_t)Nn * HEADS);
  float* kh  = carve(2 * HEADS);
  (void)ws_size; (void)n_in;

  const int* srcI = eidx;
  const int* dstI = eidx + E_;

  const int B256 = 256;
  dim3 gemmGrid(Nn / 16, 1);

  // x = emb_table[entity] @ W_proj + b_proj
  gemm_wmma_f32<<<gemmGrid, B256, 0, stream>>>(emb, W_proj, b_proj, entity, x, Nn, F_, DDIM);

  const long long nd = (long long)Nn * DDIM;
  const long long nh = (long long)Nn * HEADS;

  for (int l = 0; l < L_; ++l) {
    const float* W_src_l  = W_src  + (size_t)l * DDIM * DDIM;
    const float* W_edge_l = W_edge + (size_t)l * DDIM * DDIM;
    const float* att_l    = att    + (size_t)l * HEADS * 3 * CDIM;
    const float* cbias_l  = cbias  + (size_t)l * DDIM;

    // xh = x @ W_src_l
    gemm_wmma_f32<<<gemmGrid, B256, 0, stream>>>(x, W_src_l, nullptr, nullptr, xh, Nn, DDIM, DDIM);

    node_scores<<<(int)((nh + B256 - 1) / B256), B256, 0, stream>>>(xh, att_l, si, sj, Nn);
    edge_k<<<1, DDIM, 0, stream>>>(W_ew, b_ew, W_edge_l, att_l, kh);

    zero_f32<<<(int)((nh + B256 - 1) / B256), B256, 0, stream>>>(mM, nh);
    zero_f32<<<(int)((nh + B256 - 1) / B256), B256, 0, stream>>>(den, nh);
    zero_f32<<<(int)((nd + B256 - 1) / B256), B256, 0, stream>>>(acc, nd);

    int eBlocks = (Ntot + B256 - 1) / B256;
    edge_score<<<eBlocks, B256, 0, stream>>>(srcI, dstI, eweight, si, sj, kh,
                                             aE, (unsigned int*)mM, E_, Ntot);
    edge_den<<<eBlocks, B256, 0, stream>>>(dstI, aE, mM, den, E_, Ntot);

    long long aggThreads = (long long)Ntot * 32;
    edge_aggregate<<<(int)((aggThreads + B256 - 1) / B256), B256, 0, stream>>>(
        srcI, dstI, aE, mM, den, xh, acc, E_, Ntot);

    float* xnext = (l == L_ - 1) ? (float*)d_out : x;
    finalize<<<(int)((nd + B256 - 1) / B256), B256, 0, stream>>>(acc, cbias_l, xnext, nd,
                                                                 (l != L_ - 1) ? 1 : 0);
  }
}